// GAT_34359738368537
// MI455X (gfx1250) — compile-verified
//
#include <hip/hip_runtime.h>
#include <hip/hip_bf16.h>
#include <cstdint>
#include <cstddef>

#define NEG_SLOPE 0.2f
#define EPSV 1e-16f

typedef __attribute__((ext_vector_type(2))) float v2f;
typedef __attribute__((ext_vector_type(8))) float v8f;

// ---------------------------------------------------------------------------
// WMMA fp32 GEMM: C[M,Nout] = A[M,K] @ B[K,LDB] (B zero-padded to LDB cols).
// One wave computes one 16x16 output tile with V_WMMA_F32_16X16X4_F32.
// K and LDB are compile-time -> fully unrolled loop, loads get immediate
// offsets, no masking anywhere in the hot loop (EXEC all-1s at every WMMA).
// M must be a multiple of 16 (100000 = 6250*16). Store masks col < Nout.
// ---------------------------------------------------------------------------
template <int K, int LDB>
__global__ void wmma_gemm_f32(const float* __restrict__ A,
                              const float* __restrict__ B,
                              float* __restrict__ C,
                              int Nout, int ntn, int totalTiles)
{
    int wave = threadIdx.x >> 5;
    int lane = threadIdx.x & 31;
    int tile = blockIdx.x * (blockDim.x >> 5) + wave;
    if (tile >= totalTiles) return;          // wave-uniform exit

    int m0   = (tile / ntn) << 4;
    int n0   = (tile % ntn) << 4;
    int half = lane >> 4;                    // 0: lanes 0-15, 1: lanes 16-31
    int l    = lane & 15;
    int col  = n0 + l;

    // A fragment: half 0 holds K = kk+0/kk+1, half 1 holds kk+2/kk+3
    const float* ap = A + (size_t)(m0 + l) * K + 2 * half;
    // B fragment: VGPR0 = row kk+2*half, VGPR1 = row kk+2*half+1, column `col`
    const float* bp = B + (size_t)(2 * half) * LDB + col;

    v8f c = {0.f, 0.f, 0.f, 0.f, 0.f, 0.f, 0.f, 0.f};
#pragma unroll
    for (int kk = 0; kk < K; kk += 4) {
        v2f a = *(const v2f*)(ap + kk);      // 8B-aligned -> global_load_b64
        v2f b;
        b.x = bp[(size_t)kk * LDB];
        b.y = bp[(size_t)(kk + 1) * LDB];
        c = __builtin_amdgcn_wmma_f32_16x16x4_f32(
                /*neg_a=*/false, a, /*neg_b=*/false, b,
                /*c_mod=*/(short)0, c, /*reuse_a=*/false, /*reuse_b=*/false);
    }

    if (col < Nout) {
#pragma unroll
        for (int r = 0; r < 8; ++r) {
            int row = m0 + r + 8 * half;     // C VGPR r: rows r (low), r+8 (high)
            C[(size_t)row * Nout + col] = c[r];
        }
    }
}

// Zero-pad B[K,Nout] -> Bp[K,LDB]
__global__ void pad_b(const float* __restrict__ B, float* __restrict__ Bp,
                      int K, int Nout, int LDB)
{
    int t = blockIdx.x * blockDim.x + threadIdx.x;
    if (t >= K * LDB) return;
    int k = t / LDB, c = t % LDB;
    Bp[t] = (c < Nout) ? B[k * Nout + c] : 0.f;
}

// alpha_src[i] = h[i,:] . a_src ; alpha_dst[i] = h[i,:] . a_dst
__global__ void gemv_alpha(const float* __restrict__ h,
                           const float* __restrict__ a_src,
                           const float* __restrict__ a_dst,
                           float* __restrict__ as_, float* __restrict__ ad_,
                           int N, int C)
{
    int i = blockIdx.x * blockDim.x + threadIdx.x;
    if (i >= N) return;
    const float* row = h + (size_t)i * C;
    float s = 0.f, d = 0.f;
    for (int c = 0; c < C; ++c) {
        float v = row[c];
        s = fmaf(v, a_src[c], s);
        d = fmaf(v, a_dst[c], d);
    }
    as_[i] = s;
    ad_[i] = d;
}

// out[i,c] = b[c]; m[i] = -FLT_MAX; denom[i] = 0   (run every launch: deterministic)
__global__ void init_layer(float* __restrict__ m, float* __restrict__ denom,
                           float* __restrict__ out, const float* __restrict__ b,
                           int N, int Cout)
{
    long long t = (long long)blockIdx.x * blockDim.x + threadIdx.x;
    if (t >= (long long)N * Cout) return;
    int i = (int)(t / Cout);
    int c = (int)(t % Cout);
    out[t] = b[c];
    if (c == 0) { m[i] = -3.402823466e+38f; denom[i] = 0.f; }
}

// Per-edge raw score + segment max (native global_atomic_max_num_f32).
// Edges e >= E are implicit self loops (src=dst=e-E, w=1).
__global__ void edge_alpha_max(const int* __restrict__ ei, const float* __restrict__ w,
                               const float* __restrict__ as_, const float* __restrict__ ad_,
                               float* __restrict__ alpha_e, float* __restrict__ m,
                               int E, int Etot)
{
    int e = blockIdx.x * blockDim.x + threadIdx.x;
    if (e >= Etot) return;
    int s, d; float we;
    if (e < E) { s = ei[e]; d = ei[E + e]; we = w[e]; }
    else       { s = d = e - E;            we = 1.0f; }
    float al = as_[s] + ad_[d];
    al = (al > 0.f) ? al : NEG_SLOPE * al;     // leaky relu
    al += 1.0f - 1.0f / we;                    // edge-weight adjustment
    alpha_e[e] = al;
    unsafeAtomicMax(&m[d], al);
}

// ea = exp(alpha - m[dst]); denom[dst] += ea
__global__ void edge_exp_sum(const int* __restrict__ ei, const float* __restrict__ m,
                             float* __restrict__ alpha_e, float* __restrict__ denom,
                             int E, int Etot)
{
    int e = blockIdx.x * blockDim.x + threadIdx.x;
    if (e >= Etot) return;
    int d = (e < E) ? ei[E + e] : (e - E);
    float ea = expf(alpha_e[e] - m[d]);
    alpha_e[e] = ea;
    unsafeAtomicAdd(&denom[d], ea);
}

// out[dst, c] += h[src, c] * ea / (denom[dst] + eps); one thread per (edge, channel)
// CoutPad = 1<<Clog2 (64 for layer 1, 8 for layer 2's Cout=7)
__global__ void edge_aggregate(const int* __restrict__ ei,
                               const float* __restrict__ alpha_e,
                               const float* __restrict__ denom,
                               const float* __restrict__ h,
                               float* __restrict__ out,
                               int E, int Etot, int Cout, int Clog2)
{
    long long t = (long long)blockIdx.x * blockDim.x + threadIdx.x;
    int e = (int)(t >> Clog2);
    int c = (int)(t & ((1 << Clog2) - 1));
    if (e >= Etot || c >= Cout) return;
    int s, d;
    if (e < E) { s = ei[e]; d = ei[E + e]; }
    else       { s = d = e - E; }
    float al = alpha_e[e] / (denom[d] + EPSV);
    unsafeAtomicAdd(&out[(size_t)d * Cout + c], h[(size_t)s * Cout + c] * al);
}

__global__ void relu_k(float* __restrict__ x, long long n)
{
    long long i = (long long)blockIdx.x * blockDim.x + threadIdx.x;
    if (i < n) x[i] = fmaxf(x[i], 0.f);
}

static inline int cdiv(long long a, long long b) { return (int)((a + b - 1) / b); }

extern "C" void kernel_launch(void* const* d_in, const int* in_sizes, int n_in,
                              void* d_out, int out_size, void* d_ws, size_t ws_size,
                              hipStream_t stream)
{
    (void)n_in; (void)out_size; (void)ws_size;

    const float* x    = (const float*)d_in[0];
    const int*   ei   = (const int*)  d_in[1];   // [2, E] row-major: src then dst
    const float* w    = (const float*)d_in[2];
    const float* W1   = (const float*)d_in[3];   // [128, 64]
    const float* as1  = (const float*)d_in[4];
    const float* ad1  = (const float*)d_in[5];
    const float* b1   = (const float*)d_in[6];
    const float* W2   = (const float*)d_in[7];   // [64, 7]
    const float* as2  = (const float*)d_in[8];
    const float* ad2  = (const float*)d_in[9];
    const float* b2   = (const float*)d_in[10];

    const int N    = in_sizes[0] / 128;   // 100000
    const int E    = in_sizes[2];         // 1600000
    const int Etot = E + N;               // + self loops
    float* out = (float*)d_out;           // [N, 7]

    // workspace layout (fp32)
    float* ws  = (float*)d_ws;
    float* h1  = ws;                          // N*64
    float* o1  = h1  + (size_t)N * 64;        // N*64
    float* asv = o1  + (size_t)N * 64;        // N
    float* adv = asv + N;                     // N
    float* mx  = adv + N;                     // N
    float* dn  = mx  + N;                     // N
    float* ae  = dn  + N;                     // Etot
    float* h2  = ae  + Etot;                  // N*7
    float* W2p = h2  + (size_t)N * 7;         // 64*16 zero-padded W2

    const int TB = 256;                        // 8 waves per block
    const int wavesPerBlock = TB / 32;

    // ---------------- Layer 1 ----------------
    {
        int ntn = 64 / 16;                                   // 4 N-tiles
        int tiles = (N / 16) * ntn;                          // 25000
        wmma_gemm_f32<128, 64><<<cdiv(tiles, wavesPerBlock), TB, 0, stream>>>(
            x, W1, h1, 64, ntn, tiles);

        gemv_alpha<<<cdiv(N, TB), TB, 0, stream>>>(h1, as1, ad1, asv, adv, N, 64);
        init_layer<<<cdiv((long long)N * 64, TB), TB, 0, stream>>>(mx, dn, o1, b1, N, 64);
        edge_alpha_max<<<cdiv(Etot, TB), TB, 0, stream>>>(ei, w, asv, adv, ae, mx, E, Etot);
        edge_exp_sum<<<cdiv(Etot, TB), TB, 0, stream>>>(ei, mx, ae, dn, E, Etot);
        edge_aggregate<<<cdiv((long long)Etot << 6, TB), TB, 0, stream>>>(
            ei, ae, dn, h1, o1, E, Etot, 64, 6);
        relu_k<<<cdiv((long long)N * 64, TB), TB, 0, stream>>>(o1, (long long)N * 64);
    }

    // ---------------- Layer 2 ----------------
    {
        pad_b<<<cdiv(64 * 16, TB), TB, 0, stream>>>(W2, W2p, 64, 7, 16);

        int ntn = 1;                                         // Nout=7 -> one masked tile
        int tiles = N / 16;                                  // 6250
        wmma_gemm_f32<64, 16><<<cdiv(tiles, wavesPerBlock), TB, 0, stream>>>(
            o1, W2p, h2, 7, ntn, tiles);

        gemv_alpha<<<cdiv(N, TB), TB, 0, stream>>>(h2, as2, ad2, asv, adv, N, 7);
        init_layer<<<cdiv((long long)N * 7, TB), TB, 0, stream>>>(mx, dn, out, b2, N, 7);
        edge_alpha_max<<<cdiv(Etot, TB), TB, 0, stream>>>(ei, w, asv, adv, ae, mx, E, Etot);
        edge_exp_sum<<<cdiv(Etot, TB), TB, 0, stream>>>(ei, mx, ae, dn, E, Etot);
        edge_aggregate<<<cdiv((long long)Etot << 3, TB), TB, 0, stream>>>(
            ei, ae, dn, h2, out, E, Etot, 7, 3);
    }
}